// DifferentiableTopKSelector_16621523435814
// MI455X (gfx1250) — compile-verified
//
#include <hip/hip_runtime.h>
#include <stdint.h>

#define COLS 8192
#define TPB  256
#define NW   (TPB / 32)
#define EPT  (COLS / TPB)   // 32 elements per thread
#define KSEL 32
#define EPSV 1e-10f
#define MSUB 10000.0f

// CDNA5 async global->LDS DMA path (ASYNCcnt), guarded so compile never breaks.
#if __has_builtin(__builtin_amdgcn_global_load_async_to_lds_b128)
#define USE_ASYNC_LDS 1
#else
#define USE_ASYNC_LDS 0
#endif

#if USE_ASYNC_LDS
typedef int v4i_t __attribute__((vector_size(16)));
typedef __attribute__((address_space(1))) v4i_t* g_v4i_ptr;   // global
typedef __attribute__((address_space(3))) v4i_t* l_v4i_ptr;   // LDS
typedef __attribute__((address_space(3))) char*  l_chr_ptr;
#endif

__device__ __forceinline__ float waveReduceMax(float v) {
#pragma unroll
  for (int off = 16; off > 0; off >>= 1)
    v = fmaxf(v, __shfl_xor(v, off, 32));
  return v;
}

__device__ __forceinline__ float waveReduceSum(float v) {
#pragma unroll
  for (int off = 16; off > 0; off >>= 1)
    v += __shfl_xor(v, off, 32);
  return v;
}

__global__ __launch_bounds__(TPB) void soft_topk_fused(
    const float* __restrict__ scores,
    const float* __restrict__ u,
    float* __restrict__ out) {
  __shared__ float lds_row[COLS];   // staged scores row (32 KB of the 320 KB WGP LDS)
  __shared__ float redf[NW];
  __shared__ int   redi[NW];

  const int tid  = threadIdx.x;
  const int lane = tid & 31;
  const int wid  = tid >> 5;
  const size_t rowBase = (size_t)blockIdx.x * COLS;
  const float* __restrict__ srow = scores + rowBase;
  const float* __restrict__ urow = u + rowBase;
  float* __restrict__ orow = out + rowBase;

  // ---- Stage scores row into LDS; on CDNA5 use async DMA, overlapped with gumbel math
#if USE_ASYNC_LDS
#pragma unroll
  for (int c = 0; c < (COLS * 4) / (TPB * 16); ++c) {   // 8 x b128 per thread
    const int boff = (c * TPB + tid) * 16;
    g_v4i_ptr gsrc = (g_v4i_ptr)((uintptr_t)srow + (uintptr_t)boff);
    l_v4i_ptr ldst = (l_v4i_ptr)((l_chr_ptr)lds_row + boff);
    __builtin_amdgcn_global_load_async_to_lds_b128(gsrc, ldst, 0, 0);
  }
#else
#pragma unroll
  for (int j = 0; j < EPT; ++j) {
    const int c = tid + j * TPB;
    lds_row[c] = srow[c];
  }
#endif

  __builtin_prefetch(urow + tid, 0, 1);   // global_prefetch_b8

  // Gumbel noise from u while the DMA is in flight (u read exactly once -> NT load)
  float rem[EPT], soft[EPT];
#pragma unroll
  for (int j = 0; j < EPT; ++j) {
    const float uv = __builtin_nontemporal_load(urow + tid + j * TPB);
    rem[j]  = -__logf(-__logf(uv + EPSV) + EPSV);
    soft[j] = 0.0f;
  }

#if USE_ASYNC_LDS
#if __has_builtin(__builtin_amdgcn_s_wait_asynccnt)
  __builtin_amdgcn_s_wait_asynccnt(0);
#else
  asm volatile("s_wait_asynccnt 0" ::: "memory");
#endif
#endif
  __syncthreads();

  // perturbed = scores + gumbel   (TEMP == 1)
#pragma unroll
  for (int j = 0; j < EPT; ++j)
    rem[j] += lds_row[tid + j * TPB];

  // ---- Hard top-K: 32 iterative argmaxes over the LDS score copy (destroys it)
  unsigned hard = 0u;
  for (int it = 0; it < KSEL; ++it) {
    float v = -__builtin_inff();
    int ix = COLS;
#pragma unroll
    for (int j = 0; j < EPT; ++j) {
      const int c = tid + j * TPB;
      const float x = lds_row[c];
      if (x > v || (x == v && c < ix)) { v = x; ix = c; }
    }
#pragma unroll
    for (int off = 16; off > 0; off >>= 1) {   // wave32 argmax (tie -> lowest index)
      const float v2 = __shfl_xor(v, off, 32);
      const int   i2 = __shfl_xor(ix, off, 32);
      if (v2 > v || (v2 == v && i2 < ix)) { v = v2; ix = i2; }
    }
    if (lane == 0) { redf[wid] = v; redi[wid] = ix; }
    __syncthreads();
    // every thread folds the 8 wave winners -> all agree on block winner bi
    float bv = redf[0];
    int   bi = redi[0];
#pragma unroll
    for (int w = 1; w < NW; ++w)
      if (redf[w] > bv || (redf[w] == bv && redi[w] < bi)) { bv = redf[w]; bi = redi[w]; }
    if ((bi & (TPB - 1)) == tid) hard |= 1u << (bi >> 8);   // owner records its bit
    if (tid == 0) lds_row[bi] = -__builtin_inff();          // knock out the winner
    __syncthreads();
  }

  // ---- One-time stable shift: softmax is shift-invariant, and suppression only
  //      lowers values, so the iteration-0 row max bounds all later iterations.
  float c0;
  {
    float m = -__builtin_inff();
#pragma unroll
    for (int j = 0; j < EPT; ++j) m = fmaxf(m, rem[j]);
    m = waveReduceMax(m);
    if (lane == 0) redf[wid] = m;
    __syncthreads();
    float mm = redf[0];
#pragma unroll
    for (int w = 1; w < NW; ++w) mm = fmaxf(mm, redf[w]);
    c0 = mm;
    __syncthreads();
  }

  // ---- Soft top-K: 32 rounds of row softmax + suppression, one reduction/round
  for (int it = 0; it < KSEL; ++it) {
    float e[EPT];
    float ps = 0.0f;
#pragma unroll
    for (int j = 0; j < EPT; ++j) {
      const float ev = __expf(rem[j] - c0);   // v_exp_f32 — the dominant cost
      e[j] = ev;
      ps += ev;
    }
    ps = waveReduceSum(ps);
    if (lane == 0) redf[wid] = ps;
    __syncthreads();
    float total = redf[0];
#pragma unroll
    for (int w = 1; w < NW; ++w) total += redf[w];
    __syncthreads();
    const float inv = __builtin_amdgcn_rcpf(total);
#pragma unroll
    for (int j = 0; j < EPT; ++j) {
      const float sij = e[j] * inv;
      soft[j] += sij;
      rem[j] = __builtin_fmaf(sij, -MSUB, rem[j]);   // remaining -= s * 10000
    }
  }

  // ---- Straight-through combine; streaming NT store
#pragma unroll
  for (int j = 0; j < EPT; ++j) {
    const float h = ((hard >> j) & 1u) ? 1.0f : 0.0f;
    __builtin_nontemporal_store((h - soft[j]) + soft[j], orow + tid + j * TPB);
  }
}

extern "C" void kernel_launch(void* const* d_in, const int* in_sizes, int n_in,
                              void* d_out, int out_size, void* d_ws, size_t ws_size,
                              hipStream_t stream) {
  (void)n_in; (void)d_ws; (void)ws_size; (void)in_sizes;
  const float* scores = (const float*)d_in[0];
  const float* u      = (const float*)d_in[1];
  float* out          = (float*)d_out;
  const int rows = out_size / COLS;           // 4096
  soft_topk_fused<<<dim3(rows), dim3(TPB), 0, stream>>>(scores, u, out);
}